// Net_46265387712704
// MI455X (gfx1250) — compile-verified
//
#include <hip/hip_runtime.h>
#include <math.h>

typedef __attribute__((ext_vector_type(2))) float v2f;
typedef __attribute__((ext_vector_type(8))) float v8f;

// ---------------------------------------------------------------------------
// Dual projection of an [N x K] f32 matrix by two [K x 16] weights using
// V_WMMA_F32_16X16X4_F32. One wave handles a 16-row tile; A fragment shared
// between the two accumulators. Bias is folded into the root projection.
// A layout (16x4 f32): lane l -> row M=l%16; VGPR0 = K=(l/16)*2, VGPR1 = K+1.
// B layout (4x16 f32): lane l -> col N=l%16; VGPR0 = K=(l/16)*2, VGPR1 = K+1.
// C/D (16x16 f32, 8 VGPRs): VGPR i, lanes 0-15 -> M=i, lanes 16-31 -> M=i+8.
// ---------------------------------------------------------------------------
template <int K>
__global__ __launch_bounds__(256) void proj16_wmma(
    const float* __restrict__ X, const float* __restrict__ Wrel,
    const float* __restrict__ Wroot, const float* __restrict__ bias,
    float* __restrict__ out_rel, float* __restrict__ out_root, int n_tiles) {
  const int wave = threadIdx.x >> 5;
  const int lane = threadIdx.x & 31;
  const int tile = blockIdx.x * (blockDim.x >> 5) + wave;
  if (tile >= n_tiles) return;  // wave-uniform: EXEC stays all-ones for WMMA

  const int col = lane & 15;
  const int hi = lane >> 4;
  const int kb = hi << 1;  // 0 or 2
  const int row0 = tile * 16;

  v8f acc_rel = {};
  v8f acc_root = {};

  const float* xrow = X + (size_t)(row0 + col) * K;

#pragma unroll 4
  for (int k0 = 0; k0 < K; k0 += 4) {
    v2f a, br, bo;
    a.x = xrow[k0 + kb];
    a.y = xrow[k0 + kb + 1];
    br.x = Wrel[(k0 + kb) * 16 + col];
    br.y = Wrel[(k0 + kb + 1) * 16 + col];
    bo.x = Wroot[(k0 + kb) * 16 + col];
    bo.y = Wroot[(k0 + kb + 1) * 16 + col];
    acc_rel = __builtin_amdgcn_wmma_f32_16x16x4_f32(false, a, false, br,
                                                    (short)0, acc_rel, false,
                                                    false);
    acc_root = __builtin_amdgcn_wmma_f32_16x16x4_f32(false, a, false, bo,
                                                     (short)0, acc_root, false,
                                                     false);
  }

  const float bv = bias[col];
  const int rbase = row0 + (hi << 3);
#pragma unroll
  for (int i = 0; i < 8; ++i) {
    out_rel[(size_t)(rbase + i) * 16 + col] = acc_rel[i];
    out_root[(size_t)(rbase + i) * 16 + col] = acc_root[i] + bv;
  }
}

// ---------------------------------------------------------------------------
// Edge scatter: one thread per (edge, feature). 16 consecutive threads share
// an edge -> coalesced 64B gather from the L2-resident projected table and
// coalesced hardware f32 atomics into the accumulator.
// ---------------------------------------------------------------------------
__global__ __launch_bounds__(256) void edge_scatter(
    const long long* __restrict__ src, const long long* __restrict__ dst,
    const float* __restrict__ vals, float* __restrict__ agg, int n_edges) {
  long long tid = (long long)blockIdx.x * blockDim.x + threadIdx.x;
  if (tid >= (long long)n_edges * 16) return;
  int e = (int)(tid >> 4);
  int f = (int)(tid & 15);
  int s = (int)src[e];
  int d = (int)dst[e];
  float v = vals[(size_t)s * 16 + f];
  __hip_atomic_fetch_add(&agg[(size_t)d * 16 + f], v, __ATOMIC_RELAXED,
                         __HIP_MEMORY_SCOPE_AGENT);
}

// h = relu(agg + root)   (bias already folded into root). float4 vectorized.
__global__ __launch_bounds__(256) void relu_combine(
    const float* __restrict__ agg, const float* __restrict__ root,
    float* __restrict__ h, int n4) {
  int i = blockIdx.x * blockDim.x + threadIdx.x;
  if (i >= n4) return;
  float4 a = ((const float4*)agg)[i];
  float4 r = ((const float4*)root)[i];
  float4 o;
  o.x = fmaxf(a.x + r.x, 0.0f);
  o.y = fmaxf(a.y + r.y, 0.0f);
  o.z = fmaxf(a.z + r.z, 0.0f);
  o.w = fmaxf(a.w + r.w, 0.0f);
  ((float4*)h)[i] = o;
}

// out = log_softmax(agg + root) over 16 classes; one thread per row.
__global__ __launch_bounds__(256) void logsoftmax16(
    const float* __restrict__ agg, const float* __restrict__ root,
    float* __restrict__ out, int n_rows) {
  int r = blockIdx.x * blockDim.x + threadIdx.x;
  if (r >= n_rows) return;
  float v[16];
  float m = -INFINITY;
#pragma unroll
  for (int j = 0; j < 16; ++j) {
    v[j] = agg[(size_t)r * 16 + j] + root[(size_t)r * 16 + j];
    m = fmaxf(m, v[j]);
  }
  float s = 0.0f;
#pragma unroll
  for (int j = 0; j < 16; ++j) s += expf(v[j] - m);
  float ls = logf(s);
#pragma unroll
  for (int j = 0; j < 16; ++j) out[(size_t)r * 16 + j] = v[j] - m - ls;
}

extern "C" void kernel_launch(void* const* d_in, const int* in_sizes, int n_in,
                              void* d_out, int out_size, void* d_ws,
                              size_t ws_size, hipStream_t stream) {
  const float* x = (const float*)d_in[0];
  const long long* ei = (const long long*)d_in[1];  // int64 per reference
  const float* w1_rel = (const float*)d_in[2];
  const float* w1_root = (const float*)d_in[3];
  const float* b1 = (const float*)d_in[4];
  const float* w2_rel = (const float*)d_in[5];
  const float* w2_root = (const float*)d_in[6];
  const float* b2 = (const float*)d_in[7];
  float* out = (float*)d_out;

  const int N = in_sizes[0] / 128;  // 100000
  const int E = in_sizes[1] / 2;    // 1600000
  const long long* src = ei;
  const long long* dst = ei + E;

  const size_t fsz = (size_t)N * 16;
  float* buf_rel = (float*)d_ws;       // projected-by-W_rel table / reused L2
  float* buf_root = buf_rel + fsz;     // root projection (+bias)
  float* buf_agg = buf_root + fsz;     // scatter accumulator
  float* buf_h = buf_agg + fsz;        // layer-1 activations

  const int n_tiles = N / 16;  // N is a multiple of 16 (100000/16 = 6250)
  dim3 pb(256);
  dim3 pg((n_tiles + 7) / 8);  // 8 waves (tiles) per block

  const long long work = (long long)E * 16;
  const int sblocks = (int)((work + 255) / 256);
  const int n4 = (int)(fsz / 4);

  // ---- Layer 1: project first (linearity of segment_sum), then scatter ----
  proj16_wmma<128><<<pg, pb, 0, stream>>>(x, w1_rel, w1_root, b1, buf_rel,
                                          buf_root, n_tiles);
  hipMemsetAsync(buf_agg, 0, fsz * sizeof(float), stream);
  edge_scatter<<<sblocks, 256, 0, stream>>>(src, dst, buf_rel, buf_agg, E);
  relu_combine<<<(n4 + 255) / 256, 256, 0, stream>>>(buf_agg, buf_root, buf_h,
                                                     n4);

  // ---- Layer 2 ----
  proj16_wmma<16><<<pg, pb, 0, stream>>>(buf_h, w2_rel, w2_root, b2, buf_rel,
                                         buf_root, n_tiles);
  hipMemsetAsync(buf_agg, 0, fsz * sizeof(float), stream);
  edge_scatter<<<sblocks, 256, 0, stream>>>(src, dst, buf_rel, buf_agg, E);
  logsoftmax16<<<(N + 255) / 256, 256, 0, stream>>>(buf_agg, buf_root, out, N);
}